// GAT_36644660970168
// MI455X (gfx1250) — compile-verified
//
#include <hip/hip_runtime.h>
#include <hip/hip_bf16.h>

typedef float v2f __attribute__((ext_vector_type(2)));
typedef float v8f __attribute__((ext_vector_type(8)));

#define TT 8
#define FF 64
#define HC 128   // H*C
#define CC 64

// ---------------- utility: order-preserving float<->uint for atomic max ----
__device__ __forceinline__ unsigned f2ord(float f) {
    unsigned b = __float_as_uint(f);
    return (b & 0x80000000u) ? ~b : (b | 0x80000000u);
}
__device__ __forceinline__ float ord2f(unsigned u) {
    unsigned b = (u & 0x80000000u) ? (u ^ 0x80000000u) : ~u;
    return __uint_as_float(b);
}
__device__ __forceinline__ float lrelu(float v) { return v > 0.f ? v : 0.2f * v; }

// ---------------- zero init ------------------------------------------------
__global__ void zero_kernel(unsigned* __restrict__ p, long n) {
    long i = (long)blockIdx.x * blockDim.x + threadIdx.x;
    if (i < n) p[i] = 0u;
}

// ---------------- GEMM: xl[t][n][hc] = X[m=n*8+t][f] @ W[f][hc] ------------
// One 16x16 output tile per wave; K=64 via 16 x v_wmma_f32_16x16x4_f32.
// Block = 256 threads = 8 waves = the 8 column tiles of one row tile.
__global__ void gemm_xl_wmma(const float* __restrict__ X,
                             const float* __restrict__ Wm,
                             float* __restrict__ xl,
                             long M, int N) {
    const int wave = threadIdx.x >> 5;     // ctile 0..7
    const int lane = threadIdx.x & 31;
    const long mtile = blockIdx.x;
    const int kh   = lane >> 4;            // 0: K pair {k,k+1} | 1: {k+2,k+3}
    const int lane16 = lane & 15;

    long row = mtile * 16 + lane16;        // A-matrix row for this lane
    long rowL = row < M ? row : (M - 1);   // clamp (M=400000 divisible by 16)
    const int col = wave * 16 + lane16;    // global output column (0..127)

    const float* __restrict__ Xr = X + rowL * FF;

    v8f acc = {};
#pragma unroll
    for (int k = 0; k < FF; k += 4) {
        v2f a, b;
        const int ka = k + 2 * kh;
        a.x = Xr[ka];
        a.y = Xr[ka + 1];
        b.x = Wm[(long)ka * HC + col];
        b.y = Wm[(long)(ka + 1) * HC + col];
        // D = A(16x4,f32) * B(4x16,f32) + C ; full f32 accumulate
        acc = __builtin_amdgcn_wmma_f32_16x16x4_f32(
            false, a, false, b, (short)0, acc, false, false);
    }

    // C/D layout: VGPR r -> (Mrow = r + 8*kh, Ncol = lane&15)
#pragma unroll
    for (int r = 0; r < 8; ++r) {
        long m = mtile * 16 + r + 8 * kh;
        if (m < M) {
            int n = (int)(m >> 3);         // m = n*8 + t
            int t = (int)(m & 7);
            xl[((long)t * N + n) * HC + col] = acc[r];
        }
    }
}

// ---------------- per-node attention scores --------------------------------
__global__ void scores_kernel(const float* __restrict__ xl,
                              const float* __restrict__ att_src,
                              const float* __restrict__ att_dst,
                              float* __restrict__ aS, float* __restrict__ aD,
                              int N) {
    __shared__ float s_as[HC], s_ad[HC];
    if (threadIdx.x < HC) {
        s_as[threadIdx.x] = att_src[threadIdx.x];
        s_ad[threadIdx.x] = att_dst[threadIdx.x];
    }
    __syncthreads();
    int n = blockIdx.x * blockDim.x + threadIdx.x;
    int t = blockIdx.y;
    if (n >= N) return;
    const float* __restrict__ row = xl + ((long)t * N + n) * HC;
    float as0 = 0.f, as1 = 0.f, ad0 = 0.f, ad1 = 0.f;
#pragma unroll 8
    for (int c = 0; c < CC; ++c) {
        float x0 = row[c], x1 = row[CC + c];
        as0 += x0 * s_as[c];       ad0 += x0 * s_ad[c];
        as1 += x1 * s_as[CC + c];  ad1 += x1 * s_ad[CC + c];
    }
    long base = ((long)t * N + n) * 2;
    aS[base] = as0; aS[base + 1] = as1;
    aD[base] = ad0; aD[base + 1] = ad1;
}

// ---------------- softmax pass 1: per-dst max ------------------------------
__global__ void edge_max_kernel(const int* __restrict__ src, const int* __restrict__ dst,
                                const float* __restrict__ aS, const float* __restrict__ aD,
                                unsigned* __restrict__ emax, int E, int N) {
    int e = blockIdx.x * blockDim.x + threadIdx.x;
    int t = blockIdx.y;
    if (e >= E) return;
    int s = src[e], d = dst[e];
    long bs = ((long)t * N + s) * 2, bd = ((long)t * N + d) * 2;
#pragma unroll
    for (int h = 0; h < 2; ++h) {
        float v = lrelu(aS[bs + h] + aD[bd + h]);
        atomicMax(&emax[bd + h], f2ord(v));
    }
}

// ---------------- softmax pass 2: sum of exp -------------------------------
__global__ void edge_expsum_kernel(const int* __restrict__ src, const int* __restrict__ dst,
                                   const float* __restrict__ aS, const float* __restrict__ aD,
                                   const unsigned* __restrict__ emax,
                                   float* __restrict__ denom, int E, int N) {
    int e = blockIdx.x * blockDim.x + threadIdx.x;
    int t = blockIdx.y;
    if (e >= E) return;
    int s = src[e], d = dst[e];
    long bs = ((long)t * N + s) * 2, bd = ((long)t * N + d) * 2;
#pragma unroll
    for (int h = 0; h < 2; ++h) {
        float v = lrelu(aS[bs + h] + aD[bd + h]);
        atomicAdd(&denom[bd + h], __expf(v - ord2f(emax[bd + h])));
    }
}

// ---------------- pass 3: alpha-weighted scatter (head mean fused) ---------
// 64 lanes per (edge,t); each lane owns one channel c, handles both heads.
__global__ void edge_aggregate_kernel(const int* __restrict__ src, const int* __restrict__ dst,
                                      const float* __restrict__ aS, const float* __restrict__ aD,
                                      const unsigned* __restrict__ emax,
                                      const float* __restrict__ denom,
                                      const float* __restrict__ xl,
                                      float* __restrict__ out, int E, int N) {
    int slot = threadIdx.x >> 6;           // 4 edges per 256-thread block
    int c    = threadIdx.x & 63;
    long e = (long)blockIdx.x * 4 + slot;
    int t = blockIdx.y;
    if (e >= E) return;
    int s = src[e], d = dst[e];
    long bs = ((long)t * N + s) * 2, bd = ((long)t * N + d) * 2;

    float v0 = lrelu(aS[bs]     + aD[bd]);
    float v1 = lrelu(aS[bs + 1] + aD[bd + 1]);
    float alpha0 = __expf(v0 - ord2f(emax[bd]))     / denom[bd];
    float alpha1 = __expf(v1 - ord2f(emax[bd + 1])) / denom[bd + 1];

    const float* __restrict__ row = xl + ((long)t * N + s) * HC;
    float val = 0.5f * (alpha0 * row[c] + alpha1 * row[CC + c]);
    atomicAdd(&out[((long)d * TT + t) * CC + c], val);   // out[n][t][c]
}

// ---------------- final: bias + ELU in place -------------------------------
__global__ void bias_elu_kernel(float* __restrict__ out, const float* __restrict__ bias,
                                long total) {
    long i = (long)blockIdx.x * blockDim.x + threadIdx.x;
    if (i >= total) return;
    float v = out[i] + bias[i & 63];
    out[i] = v > 0.f ? v : (__expf(v) - 1.f);
}

// ---------------- host side ------------------------------------------------
extern "C" void kernel_launch(void* const* d_in, const int* in_sizes, int n_in,
                              void* d_out, int out_size, void* d_ws, size_t ws_size,
                              hipStream_t stream) {
    const float* h_feat   = (const float*)d_in[0];   // [1, N, T, F] -> rows m=n*8+t are contiguous 64-f32
    const int*   src      = (const int*)d_in[1];
    const int*   dst      = (const int*)d_in[2];
    const float* Wm       = (const float*)d_in[3];   // [F, H, C] = [64,128]
    const float* att_src  = (const float*)d_in[4];
    const float* att_dst  = (const float*)d_in[5];
    const float* bias     = (const float*)d_in[6];

    const int N = in_sizes[0] / (TT * FF);
    const int E = in_sizes[1];
    const long M = (long)N * TT;                     // GEMM rows

    // workspace layout (floats/uints)
    float*    xl    = (float*)d_ws;                  // [T][N][128]
    float*    aS    = xl + M * HC;                   // [T][N][2]
    float*    aD    = aS + M * 2;                    // [T][N][2]
    unsigned* emax  = (unsigned*)(aD + M * 2);       // [T][N][2]   ordered-uint, 0 == -inf
    float*    denom = (float*)(emax + M * 2);        // [T][N][2]

    // 1) zero d_out accumulator; zero emax+denom (contiguous 4*M words)
    {
        long w = M * CC;
        zero_kernel<<<(unsigned)((w + 255) / 256), 256, 0, stream>>>((unsigned*)d_out, w);
        long w2 = M * 4;
        zero_kernel<<<(unsigned)((w2 + 255) / 256), 256, 0, stream>>>(emax, w2);
    }

    // 2) WMMA GEMM: one block per 16-row tile, 8 waves = 8 column tiles
    {
        unsigned mtiles = (unsigned)((M + 15) / 16);
        gemm_xl_wmma<<<mtiles, 256, 0, stream>>>(h_feat, Wm, xl, M, N);
    }

    // 3) attention scores
    {
        dim3 g((N + 255) / 256, TT);
        scores_kernel<<<g, 256, 0, stream>>>(xl, att_src, att_dst, aS, aD, N);
    }

    // 4) segment softmax: max, then sum-of-exp
    {
        dim3 g((E + 255) / 256, TT);
        edge_max_kernel<<<g, 256, 0, stream>>>(src, dst, aS, aD, emax, E, N);
        edge_expsum_kernel<<<g, 256, 0, stream>>>(src, dst, aS, aD, emax, denom, E, N);
    }

    // 5) weighted scatter-sum (head mean fused via 0.5 factor)
    {
        dim3 g((E + 3) / 4, TT);
        edge_aggregate_kernel<<<g, 256, 0, stream>>>(src, dst, aS, aD, emax, denom, xl,
                                                     (float*)d_out, E, N);
    }

    // 6) bias + ELU in place
    {
        long total = M * CC;
        bias_elu_kernel<<<(unsigned)((total + 255) / 256), 256, 0, stream>>>(
            (float*)d_out, bias, total);
    }
}